// m1_33002528702903
// MI455X (gfx1250) — compile-verified
//
#include <hip/hip_runtime.h>

// GEMV formulation of the scalar linear recurrence:
//   out[row] = w * ( b * sum_t a^(T-1-t) x[row,t] + c * (1-a^T)/(1-a) ) + e
// Memory-bound: 128 MB single-pass stream -> ~5.5 us floor @ 23.3 TB/s.
// Reduction on the native fp32 matrix pipe V_WMMA_F32_16X16X4_F32 with the
// per-t weights broadcast across all 16 B-columns; column 0 (lanes 0/16)
// carries the 16 row partials.
//
// Timestep -> (wmma, k) slot mapping is chosen so each lane issues 16-byte
// global_load_b128s on its own row:
//   lane m    (khi=0): x[row, tb+0..3]   -> k{0,1} of wmma0, k{0,1}' of wmma1
//   lane m+16 (khi=1): x[row, tb+4..7]   -> k{2,3} of wmma0, k{2,3}' of wmma1
// The weight vector is permuted identically (pure relabeling of the sum).

typedef __attribute__((ext_vector_type(2))) float v2f;
typedef __attribute__((ext_vector_type(8))) float v8f;

#define WAVES_PER_WG 16
#define WG_THREADS   (WAVES_PER_WG * 32)

static __device__ __forceinline__ v8f wmma_f32(v2f a, v2f b, v8f c) {
    return __builtin_amdgcn_wmma_f32_16x16x4_f32(false, a, false, b,
                                                 (short)0, c, false, false);
}

__global__ __launch_bounds__(WG_THREADS)
void m1_recur_gemv_wmma(const float* __restrict__ x,
                        const float* __restrict__ pa, const float* __restrict__ pb,
                        const float* __restrict__ pc, const float* __restrict__ pw,
                        const float* __restrict__ pe,
                        float* __restrict__ out, int B, int T)
{
    const float a    = pa[0];
    const float bco  = pb[0];
    const float cco  = pc[0];
    const float wco  = pw[0];
    const float eco  = pe[0];
    const float la2  = __builtin_amdgcn_logf(a);      // v_log_f32 = log2(a)
    const float inva = __builtin_amdgcn_rcpf(a);      // 1/a, 1-ulp
    const float inva8 = __builtin_amdgcn_exp2f(-8.0f * la2);  // a^-8

    const int tid  = threadIdx.x;
    const int wave = tid >> 5;
    const int lane = tid & 31;
    const int m    = lane & 15;         // row within 16-row tile
    const int khi  = lane >> 4;         // which K-half this lane feeds

    const int row0 = blockIdx.x * 16;
    int row = row0 + m;
    if (row >= B) row = B - 1;          // keep EXEC all-ones for WMMA
    const float* rowp = x + (size_t)row * (size_t)T;

    // Disjoint contiguous T-slice per wave (weights are per-t, sums linear).
    const int tchunk = T / WAVES_PER_WG;        // 512 for T=8192
    const int tbeg   = wave * tchunk;
    const float* p   = rowp + tbeg + khi * 4;   // 16B-aligned lane stream

    v8f acc = {};
    float ex = (float)(T - 1 - tbeg - 4 * khi); // weight exponent of p[0]

    for (int t = 0; t < tchunk; t += 16) {
        const float4 x0 = *(const float4*)(p + t);      // steps g+0..3  (g = tb + 4*khi)
        const float4 x1 = *(const float4*)(p + t + 8);  // steps g+8..11
        __builtin_prefetch(p + t + 1024, 0, 1);         // global_prefetch_b8

        // weights for this lane's 8 timesteps, one v_exp_f32 + mul chain
        const float w0 = __builtin_amdgcn_exp2f(la2 * ex);
        const float w1 = w0 * inva;
        const float w2 = w1 * inva;
        const float w3 = w2 * inva;
        const float w4 = w0 * inva8;
        const float w5 = w4 * inva;
        const float w6 = w5 * inva;
        const float w7 = w6 * inva;
        ex -= 16.0f;

        v2f a0 = {x0.x, x0.y}, b0 = {w0, w1};
        v2f a1 = {x0.z, x0.w}, b1 = {w2, w3};
        v2f a2 = {x1.x, x1.y}, b2 = {w4, w5};
        v2f a3 = {x1.z, x1.w}, b3 = {w6, w7};
        acc = wmma_f32(a0, b0, acc);
        acc = wmma_f32(a1, b1, acc);
        acc = wmma_f32(a2, b2, acc);
        acc = wmma_f32(a3, b3, acc);
    }

    // Column 0 of D: lane 0 holds rows 0..7 in acc[0..7], lane 16 rows 8..15.
    __shared__ float red[WAVES_PER_WG * 16];
    if (m == 0) {
        const int base = wave * 16 + khi * 8;
        #pragma unroll
        for (int i = 0; i < 8; ++i) red[base + i] = acc[i];
    }
    __syncthreads();

    if (tid < 16) {
        float dot = 0.0f;
        #pragma unroll
        for (int wv = 0; wv < WAVES_PER_WG; ++wv) dot += red[wv * 16 + tid];
        // geometric series: sum_{j=0}^{T-1} a^j = (1 - a^T) / (1 - a)
        const float aT = __builtin_amdgcn_exp2f(la2 * (float)T);
        const float G  = (1.0f - aT) / (1.0f - a);
        const int orow = row0 + tid;
        if (orow < B) out[orow] = wco * (bco * dot + cco * G) + eco;
    }
}

extern "C" void kernel_launch(void* const* d_in, const int* in_sizes, int n_in,
                              void* d_out, int out_size, void* d_ws, size_t ws_size,
                              hipStream_t stream) {
    (void)d_ws; (void)ws_size; (void)n_in;
    const float* x  = (const float*)d_in[0];
    const float* pa = (const float*)d_in[1];
    const float* pb = (const float*)d_in[2];
    const float* pc = (const float*)d_in[3];
    const float* pw = (const float*)d_in[4];
    const float* pe = (const float*)d_in[5];
    float* out = (float*)d_out;

    const int B = out_size;                 // 4096
    const int T = in_sizes[0] / out_size;   // 8192

    const int nblocks = (B + 15) / 16;      // 256 row tiles
    m1_recur_gemv_wmma<<<nblocks, WG_THREADS, 0, stream>>>(x, pa, pb, pc, pw, pe, out, B, T);
}